// HyperGroupMix_35579509080663
// MI455X (gfx1250) — compile-verified
//
#include <hip/hip_runtime.h>
#include <hip/hip_bf16.h>
#include <cstddef>

// ---------------------------------------------------------------------------
// Problem constants: x[B=32, C=64, H=128, W=128] fp32, groups (0,16)(16,32)(32,64)
// ---------------------------------------------------------------------------
#define BB   32
#define CC   64
#define HH   128
#define WW   128
#define HWP  (HH * WW)          // 16384 pixels per channel
#define PERB 1536               // 16*16 + 16*16 + 32*32 floats of gram/whiten per batch

typedef __attribute__((ext_vector_type(2))) float v2f;
typedef __attribute__((ext_vector_type(4))) float v4f;
typedef __attribute__((ext_vector_type(8))) float v8f;

// ---------------------------------------------------------------------------
// Kernel 1: per-channel mean, unbiased var -> rsig, and lower-median-of-
// column-lower-medians (exact torch.median order statistics).
// Median via 1-bit radix select on order-preserving uint keys:
// 32 passes x 128 elems per column instead of O(n^2) counting.
// One 128-thread workgroup per (b,c); keys live in the 64KB LDS tile.
// ---------------------------------------------------------------------------
__global__ void hgm_stats_kernel(const float* __restrict__ x,
                                 float* __restrict__ mu,
                                 float* __restrict__ rsig,
                                 float* __restrict__ med) {
  __shared__ float tile[HWP];                       // 64 KB
  unsigned* keys = reinterpret_cast<unsigned*>(tile);
  const int bc = blockIdx.x;                        // 0..2047
  const int t  = threadIdx.x;                       // 0..127 (owns column w = t)
  const float* src = x + (size_t)bc * HWP;

  // fused: global read -> sums + order-preserving key store
  float s = 0.f, ss = 0.f;
  for (int i = t; i < HWP; i += 128) {
    const float v = src[i];
    s += v; ss += v * v;
    const unsigned h = __float_as_uint(v);
    keys[i] = (h & 0x80000000u) ? ~h : (h | 0x80000000u);   // monotone map
  }
  __syncthreads();

  // radix select: value of rank-63 (0-indexed) element of column t
  unsigned prefix = 0u;
  int k = 63;
  for (int bpos = 31; bpos >= 0; --bpos) {
    const unsigned target = prefix >> bpos;         // bits[31:bpos], bit bpos = 0
    int c0 = 0;
#pragma unroll 8
    for (int j = 0; j < HH; ++j)
      c0 += ((keys[j * WW + t] >> bpos) == target);
    if (k >= c0) { k -= c0; prefix |= (1u << bpos); }
  }
  __syncthreads();

  // reuse rows 0..2 of the tile for the second level + sum reduction
  keys[t]       = prefix;                           // column-median keys
  tile[128 + t] = s;
  tile[256 + t] = ss;
  __syncthreads();

  // lower median of the 128 column medians (index tie-break = torch order)
  {
    const unsigned ki = keys[t];
    int cnt = 0;
    for (int j = 0; j < 128; ++j) {
      const unsigned kj = keys[j];
      cnt += (kj < ki) || ((kj == ki) && (j < t));
    }
    if (cnt == 63) {
      const unsigned h = (ki & 0x80000000u) ? (ki & 0x7FFFFFFFu) : ~ki;
      med[bc] = __uint_as_float(h);
    }
  }

  for (int gap = 64; gap > 0; gap >>= 1) {
    if (t < gap) {
      tile[128 + t] += tile[128 + t + gap];
      tile[256 + t] += tile[256 + t + gap];
    }
    __syncthreads();
  }
  if (t == 0) {
    const float sum = tile[128], sumsq = tile[256];
    const float m   = sum / (float)HWP;
    const float var = (sumsq - sum * m) / (float)(HWP - 1);  // ddof=1
    mu[bc]   = m;
    rsig[bc] = rsqrtf(var + 1e-6f);
  }
}

// ---------------------------------------------------------------------------
// Kernel 2: gram tiles via V_WMMA_F32_16X16X4_F32.
// 6 tiles per batch: g0(16x16), g1(16x16), g2 as 2x2 of 16x16.
// 8 waves split K=16384; branch-free inner loops (diag vs off-diag decided
// once); b128 loads feed two WMMAs via k-axis permutation (valid since A and
// B share the same k mapping and K is a pure reduction).
// ---------------------------------------------------------------------------
__global__ void hgm_gram_kernel(const float* __restrict__ x,
                                float* __restrict__ gram) {
  __shared__ float red[7 * 256];       // partial C tiles from waves 1..7

  const int tileId = blockIdx.x % 6;
  const int b      = blockIdx.x / 6;
  int g, ti, tj;
  if (tileId == 0)      { g = 0; ti = 0; tj = 0; }
  else if (tileId == 1) { g = 1; ti = 0; tj = 0; }
  else                  { g = 2; ti = (tileId - 2) >> 1; tj = (tileId - 2) & 1; }
  const int cg   = (g == 2) ? 32 : 16;
  const int gs   = (g == 0) ? 0 : ((g == 1) ? 16 : 32);
  const int goff = (g == 0) ? 0 : ((g == 1) ? 256 : 512);

  const int lane = threadIdx.x & 31;
  const int wave = threadIdx.x >> 5;   // 0..7
  const int mrow = lane & 15;
  const int hi4  = 4 * (lane >> 4);    // float4 k-offset per lane half

  const float* xa = x + ((size_t)b * CC + gs + ti * 16 + mrow) * HWP + hi4;
  const float* xq = x + ((size_t)b * CC + gs + tj * 16 + mrow) * HWP + hi4;

  v8f c = {0.f, 0.f, 0.f, 0.f, 0.f, 0.f, 0.f, 0.f};
  const int k_beg = wave * (HWP / 8);          // 2048 k's per wave
  const int k_end = k_beg + (HWP / 8);

  if (ti == tj) {
#pragma unroll 4
    for (int k0 = k_beg; k0 < k_end; k0 += 8) {
      const v4f a4 = *reinterpret_cast<const v4f*>(xa + k0);
      const v2f a0 = __builtin_shufflevector(a4, a4, 0, 1);
      const v2f a1 = __builtin_shufflevector(a4, a4, 2, 3);
      c = __builtin_amdgcn_wmma_f32_16x16x4_f32(false, a0, false, a0,
                                                (short)0, c, false, false);
      c = __builtin_amdgcn_wmma_f32_16x16x4_f32(false, a1, false, a1,
                                                (short)0, c, false, false);
    }
  } else {
#pragma unroll 4
    for (int k0 = k_beg; k0 < k_end; k0 += 8) {
      const v4f a4 = *reinterpret_cast<const v4f*>(xa + k0);
      const v4f b4 = *reinterpret_cast<const v4f*>(xq + k0);
      const v2f a0 = __builtin_shufflevector(a4, a4, 0, 1);
      const v2f a1 = __builtin_shufflevector(a4, a4, 2, 3);
      const v2f b0 = __builtin_shufflevector(b4, b4, 0, 1);
      const v2f b1 = __builtin_shufflevector(b4, b4, 2, 3);
      c = __builtin_amdgcn_wmma_f32_16x16x4_f32(false, a0, false, b0,
                                                (short)0, c, false, false);
      c = __builtin_amdgcn_wmma_f32_16x16x4_f32(false, a1, false, b1,
                                                (short)0, c, false, false);
    }
  }

  if (wave > 0) {
#pragma unroll
    for (int r = 0; r < 8; ++r) red[(wave - 1) * 256 + r * 32 + lane] = c[r];
  }
  __syncthreads();
  if (wave == 0) {
    const float inv = 1.0f / ((float)HWP + 1e-6f);
#pragma unroll
    for (int r = 0; r < 8; ++r) {
      float v = c[r];
#pragma unroll
      for (int w = 0; w < 7; ++w) v += red[w * 256 + r * 32 + lane];
      v *= inv;
      const int M = r + 8 * (lane >> 4);
      const int N = lane & 15;
      const int gr = ti * 16 + M, gc = tj * 16 + N;
      if (gr == gc) v += 0.001001f;              // 0.001 + 1e-6 diag load
      gram[(size_t)b * PERB + goff + gr * cg + gc] = v;
    }
  }
}

// ---------------------------------------------------------------------------
// Kernel 3: whitening = G^{-1/2} via coupled Newton-Schulz in LDS.
// One 32x32 workgroup per (g, b) matrix (16x16 groups use a quarter).
// ---------------------------------------------------------------------------
__global__ void hgm_invsqrt_kernel(const float* __restrict__ gram,
                                   float* __restrict__ whiten) {
  __shared__ float Y[1024], Z[1024], Mm[1024];
  const int g = blockIdx.x;            // 0..2
  const int b = blockIdx.y;            // 0..31
  const int n    = (g == 2) ? 32 : 16;
  const int goff = (g == 0) ? 0 : ((g == 1) ? 256 : 512);
  const int j = threadIdx.x, i = threadIdx.y;
  const int tid = i * 32 + j;
  const bool act = (i < n) && (j < n);

  const float* A = gram + (size_t)b * PERB + goff;
  const float a = act ? A[i * n + j] : 0.f;

  Mm[tid] = a * a;
  __syncthreads();
  for (int s2 = 512; s2 > 0; s2 >>= 1) {
    if (tid < s2) Mm[tid] += Mm[tid + s2];
    __syncthreads();
  }
  const float fro = sqrtf(Mm[0] + 1e-30f);
  const float inv = 1.0f / fro;
  if (act) { Y[tid] = a * inv; Z[tid] = (i == j) ? 1.f : 0.f; }
  __syncthreads();

  for (int it = 0; it < 20; ++it) {
    float m = 0.f;
    if (act) {
      for (int k = 0; k < n; ++k) m += Z[i * 32 + k] * Y[k * 32 + j];
      m = ((i == j) ? 3.f : 0.f) - m;
    }
    __syncthreads();
    Mm[tid] = m;
    __syncthreads();
    float yn = 0.f, zn = 0.f;
    if (act) {
      for (int k = 0; k < n; ++k) {
        yn += Y[i * 32 + k] * Mm[k * 32 + j];
        zn += Mm[i * 32 + k] * Z[k * 32 + j];
      }
    }
    __syncthreads();
    if (act) { Y[tid] = 0.5f * yn; Z[tid] = 0.5f * zn; }
    __syncthreads();
  }
  if (act) whiten[(size_t)b * PERB + goff + i * n + j] = Z[tid] * rsqrtf(fro);
}

// ---------------------------------------------------------------------------
// Kernel 4: fused apply.  out = x_normed * (lm*xw_b + (1-lm)*xw_pb) + med_mix,
// where xw = W * (x - mu).  WMMA f32 16x16x4: M=out-chan, N=16 pixels, K=CG.
// Templated on CG so all operand arrays stay in registers, fully unrolled.
// ---------------------------------------------------------------------------
template <int CG>
__global__ void hgm_apply_kernel(const float* __restrict__ x,
                                 const float* __restrict__ mu,
                                 const float* __restrict__ rsig,
                                 const float* __restrict__ med,
                                 const float* __restrict__ whitenG, // + group off
                                 const float* __restrict__ lmda,
                                 const int*   __restrict__ perm,
                                 float* __restrict__ out,
                                 int gstart) {
  constexpr int KC = CG / 4;           // K chunks of 4
  constexpr int MT = CG / 16;          // 16-row output tiles
  const int b    = blockIdx.y;
  const int lane = threadIdx.x & 31;
  const int wave = threadIdx.x >> 5;   // 0..7
  const int pb   = perm[b];
  const float lm  = lmda[b];
  const float lmi = 1.0f - lm;
  const int nIdx  = lane & 15;
  const int hi    = lane >> 4;
  const int kLane = 2 * hi;

  const float* Wb = whitenG + (size_t)b  * PERB;
  const float* Wp = whitenG + (size_t)pb * PERB;
  const float* xb = x + ((size_t)b  * CC + gstart) * HWP;
  const float* xp = x + ((size_t)pb * CC + gstart) * HWP;

  // A operands (whitening rows) — reused across all pixel blocks of the wave
  v2f aB[MT][KC], aP[MT][KC];
#pragma unroll
  for (int mi = 0; mi < MT; ++mi)
#pragma unroll
    for (int kc = 0; kc < KC; ++kc) {
      const int row = mi * 16 + nIdx;
      const int kk  = kc * 4 + kLane;
      aB[mi][kc].x = Wb[row * CG + kk];
      aB[mi][kc].y = Wb[row * CG + kk + 1];
      aP[mi][kc].x = Wp[row * CG + kk];
      aP[mi][kc].y = Wp[row * CG + kk + 1];
    }
  // per-lane channel means for the B operand (fixed channel mapping per lane)
  v2f muB[KC], muP[KC];
#pragma unroll
  for (int kc = 0; kc < KC; ++kc) {
    const int ch = gstart + kc * 4 + kLane;
    muB[kc].x = mu[b * CC + ch];     muB[kc].y = mu[b * CC + ch + 1];
    muP[kc].x = mu[pb * CC + ch];    muP[kc].y = mu[pb * CC + ch + 1];
  }

  // 1024 pixel blocks of 16; 16 workgroups * 8 waves -> 8 blocks per wave
  for (int blk = blockIdx.x * 8 + wave; blk < HWP / 16; blk += gridDim.x * 8) {
    const int pix = blk * 16 + nIdx;

    v2f bBv[KC], bPv[KC];
#pragma unroll
    for (int kc = 0; kc < KC; ++kc) {
      const int ch = kc * 4 + kLane;
      bBv[kc].x = xb[(size_t)ch * HWP + pix]       - muB[kc].x;
      bBv[kc].y = xb[(size_t)(ch + 1) * HWP + pix] - muB[kc].y;
      bPv[kc].x = xp[(size_t)ch * HWP + pix]       - muP[kc].x;
      bPv[kc].y = xp[(size_t)(ch + 1) * HWP + pix] - muP[kc].y;
    }

#pragma unroll
    for (int mi = 0; mi < MT; ++mi) {
      v8f cB = {0.f, 0.f, 0.f, 0.f, 0.f, 0.f, 0.f, 0.f};
      v8f cP = {0.f, 0.f, 0.f, 0.f, 0.f, 0.f, 0.f, 0.f};
#pragma unroll
      for (int kc = 0; kc < KC; ++kc) {
        cB = __builtin_amdgcn_wmma_f32_16x16x4_f32(false, aB[mi][kc], false,
                                                   bBv[kc], (short)0, cB,
                                                   false, false);
        cP = __builtin_amdgcn_wmma_f32_16x16x4_f32(false, aP[mi][kc], false,
                                                   bPv[kc], (short)0, cP,
                                                   false, false);
      }
#pragma unroll
      for (int r = 0; r < 8; ++r) {
        const int cl  = mi * 16 + r + 8 * hi;     // C-layout row (channel)
        const int cgI = gstart + cl;
        const float gmix = lm * cB[r] + lmi * cP[r];
        const float m_  = mu[b * CC + cgI];       // hot in cache per wave
        const float rs  = rsig[b * CC + cgI];
        const float mm  = lm * med[b * CC + cgI] + lmi * med[pb * CC + cgI];
        const float xn  = (xb[(size_t)cl * HWP + pix] - m_) * rs;
        out[((size_t)b * CC + cgI) * HWP + pix] = xn * gmix + mm;
      }
    }
  }
}

// ---------------------------------------------------------------------------
extern "C" void kernel_launch(void* const* d_in, const int* in_sizes, int n_in,
                              void* d_out, int out_size, void* d_ws, size_t ws_size,
                              hipStream_t stream) {
  (void)in_sizes; (void)n_in; (void)out_size; (void)ws_size;
  const float* x    = (const float*)d_in[0];
  const float* lmda = (const float*)d_in[1];
  const int*   perm = (const int*)d_in[2];
  float* out = (float*)d_out;

  float* ws     = (float*)d_ws;
  float* mu     = ws;                       // 2048
  float* rsig   = ws + 2048;                // 2048
  float* med    = ws + 4096;                // 2048
  float* gram   = ws + 6144;                // 32*1536 = 49152
  float* whiten = ws + 6144 + BB * PERB;    // 49152

  hgm_stats_kernel<<<BB * CC, 128, 0, stream>>>(x, mu, rsig, med);
  hgm_gram_kernel<<<BB * 6, 256, 0, stream>>>(x, gram);
  hgm_invsqrt_kernel<<<dim3(3, BB), dim3(32, 32), 0, stream>>>(gram, whiten);

  hgm_apply_kernel<16><<<dim3(16, BB), 256, 0, stream>>>(
      x, mu, rsig, med, whiten + 0,   lmda, perm, out, 0);
  hgm_apply_kernel<16><<<dim3(16, BB), 256, 0, stream>>>(
      x, mu, rsig, med, whiten + 256, lmda, perm, out, 16);
  hgm_apply_kernel<32><<<dim3(16, BB), 256, 0, stream>>>(
      x, mu, rsig, med, whiten + 512, lmda, perm, out, 32);
}